// Model_65335042507145
// MI455X (gfx1250) — compile-verified
//
#include <hip/hip_runtime.h>

// Sparse GQA decode attention for MI455X (gfx1250, wave32).
//
// Memory-bound: ~1.07 GB of gathered 512B K/V rows vs ~0.5 GFLOP => roofline
// ~46us at 23.3 TB/s (about half that after skipping rows beyond nnz).
// Strategy: 1 workgroup per (b,h); scores via V_WMMA_F32_16X16X4_F32 (16
// gathered K rows x q, accumulated over 32 K-dim chunks); softmax + PV
// accumulation entirely in LDS.

#define B_    32
#define H_    32
#define HKV_  8
#define S_    8192
#define D_    128
#define LMAX_ 1024
#define GQA_  4

typedef float v2f __attribute__((ext_vector_type(2)));
typedef float v8f __attribute__((ext_vector_type(8)));

__global__ __launch_bounds__(256) void sparse_attn_kernel(
    const float* __restrict__ q,
    const float* __restrict__ K,
    const float* __restrict__ V,
    const long long* __restrict__ ind,
    const long long* __restrict__ nnz,
    float* __restrict__ out)
{
    __shared__ float s_q[D_];
    __shared__ float s_scores[LMAX_];
    __shared__ float s_out[D_];
    __shared__ float s_red[8];
    __shared__ float s_max, s_sum;

    const int bh   = blockIdx.x;
    const int tid  = threadIdx.x;
    const int wave = tid >> 5;
    const int lane = tid & 31;
    const int hi   = lane >> 4;   // which half of the wave (selects K-dim pair)
    const int r    = lane & 15;   // row within 16-row tile

    const int b   = bh / H_;
    const int h   = bh % H_;
    const int kvh = h / GQA_;

    const float* qp = q + (size_t)bh * D_;
    const float* Kb = K + (size_t)(b * HKV_ + kvh) * (size_t)S_ * D_;
    const float* Vb = V + (size_t)(b * HKV_ + kvh) * (size_t)S_ * D_;
    const long long* ip = ind + (size_t)bh * LMAX_;
    const int n = (int)nnz[bh];
    const float scale = 0.08838834764831845f;  // 1/sqrt(128)

    if (tid < D_) { s_q[tid] = qp[tid]; s_out[tid] = 0.0f; }
    __syncthreads();

    // ---------------- Phase 1: scores via WMMA f32 16x16x4 ----------------
    // Wave w owns rows [w*128, w*128+128). 8 tiles of 16 rows each.
    // A (16x4 f32): lane l holds row (l&15); VGPR0/1 hold dims 4c+2*hi, +1.
    // B (4x16 f32): q chunk replicated across all 16 columns (mirror of A
    // layout assumed: lane slot (vgpr,hi) -> K index 2*hi+vgpr). Since every
    // column is identical, D[m][*] = sum_k K[row_m][k]*q[k] in all columns.
    const int wave_base = wave * 128;
    for (int t = 0; t < 8; ++t) {
        const int base = wave_base + t * 16;
        if (base < n) {
            const int row = base + r;
            const long long idx = ip[row];            // in-bounds even if row>=n
            const float* kr = Kb + (size_t)idx * D_;
            v8f acc = {0.f, 0.f, 0.f, 0.f, 0.f, 0.f, 0.f, 0.f};
            #pragma unroll
            for (int c = 0; c < 32; ++c) {
                v2f a  = *(const v2f*)(kr + 4 * c + 2 * hi);       // 8B aligned
                v2f bq = *(const v2f*)(&s_q[4 * c + 2 * hi]);
                acc = __builtin_amdgcn_wmma_f32_16x16x4_f32(
                    false, a, false, bq, (short)0, acc, false, false);
            }
            // C/D layout: vgpr v, lanes 0-15 -> M=v; lanes 16-31 -> M=v+8.
            if (r == 0) {
                #pragma unroll
                for (int v = 0; v < 8; ++v) {
                    const int m = base + v + 8 * hi;
                    const float sc = acc[v] * scale;
                    s_scores[m] = (m < n) ? sc : -__builtin_inff();
                }
            }
        } else if (r == 0) {   // whole tile masked: skip gather + WMMA
            #pragma unroll
            for (int v = 0; v < 8; ++v)
                s_scores[base + v + 8 * hi] = -__builtin_inff();
        }
    }
    __syncthreads();

    // ---------------- Phase 2: block softmax (max, then exp+sum) ----------
    float lm = -__builtin_inff();
    for (int i = tid; i < LMAX_; i += 256) lm = fmaxf(lm, s_scores[i]);
    #pragma unroll
    for (int o = 16; o >= 1; o >>= 1) lm = fmaxf(lm, __shfl_xor(lm, o, 32));
    if (lane == 0) s_red[wave] = lm;
    __syncthreads();
    if (tid == 0) {
        float m = s_red[0];
        #pragma unroll
        for (int i = 1; i < 8; ++i) m = fmaxf(m, s_red[i]);
        s_max = m;
    }
    __syncthreads();
    const float gmax = s_max;

    float ls = 0.0f;
    for (int i = tid; i < LMAX_; i += 256) {
        const float e = __expf(s_scores[i] - gmax);   // exp(-inf)=0 for masked
        s_scores[i] = e;
        ls += e;
    }
    #pragma unroll
    for (int o = 16; o >= 1; o >>= 1) ls += __shfl_xor(ls, o, 32);
    if (lane == 0) s_red[wave] = ls;
    __syncthreads();
    if (tid == 0) {
        float ssum = 0.0f;
        #pragma unroll
        for (int i = 0; i < 8; ++i) ssum += s_red[i];
        s_sum = ssum;
    }
    __syncthreads();
    const float inv_sum = 1.0f / s_sum;

    // ---------------- Phase 3: PV accumulation -----------------------------
    // Wave w reduces its 128 rows; lane owns dims [4*lane, 4*lane+4) so each
    // V-row read is a fully-coalesced 512B wave load.
    const int d0 = lane * 4;
    float a0 = 0.f, a1 = 0.f, a2 = 0.f, a3 = 0.f;
    for (int rr = 0; rr < 128; ++rr) {
        const int l = wave_base + rr;
        if (l >= n) break;                 // wave-uniform: skip invalid tail
        const float p = s_scores[l];       // uniform LDS read
        const float* vr = Vb + (size_t)ip[l] * D_ + d0;
        const float4 vv = *(const float4*)vr;
        a0 += p * vv.x; a1 += p * vv.y; a2 += p * vv.z; a3 += p * vv.w;
    }
    atomicAdd(&s_out[d0 + 0], a0);
    atomicAdd(&s_out[d0 + 1], a1);
    atomicAdd(&s_out[d0 + 2], a2);
    atomicAdd(&s_out[d0 + 3], a3);
    __syncthreads();

    if (tid < D_) out[(size_t)bh * D_ + tid] = s_out[tid] * inv_sum;
}

extern "C" void kernel_launch(void* const* d_in, const int* in_sizes, int n_in,
                              void* d_out, int out_size, void* d_ws, size_t ws_size,
                              hipStream_t stream) {
    (void)in_sizes; (void)n_in; (void)d_ws; (void)ws_size; (void)out_size;
    const float*     q   = (const float*)d_in[0];
    const float*     K   = (const float*)d_in[1];
    const float*     V   = (const float*)d_in[2];
    const long long* ind = (const long long*)d_in[3];
    const long long* nnz = (const long long*)d_in[4];
    float* out = (float*)d_out;

    dim3 grid(B_ * H_);
    dim3 block(256);
    hipLaunchKernelGGL(sparse_attn_kernel, grid, block, 0, stream,
                       q, K, V, ind, nnz, out);
}